// VanillaGCNNet_30983894073603
// MI455X (gfx1250) — compile-verified
//
#include <hip/hip_runtime.h>
#include <hip/hip_bf16.h>

// GCN forward for MI455X (gfx1250, wave32). FP32 WMMA (v_wmma_f32_16x16x4_f32)
// for all dense transforms; atomics for the edge scatter (memory-bound part).

typedef __attribute__((ext_vector_type(2))) float v2f;
typedef __attribute__((ext_vector_type(8))) float v8f;

#define HDIM 128
#define KC 64

// ---------------------------------------------------------------------------
// Generic fp32 WMMA GEMM: C[M,Ncols] = act(A[M,K] @ B[K,Ncols] + bias)
// One 16x16 output tile per wave; 8 waves per block cover 16 rows x 128 cols.
// ---------------------------------------------------------------------------
__global__ __launch_bounds__(256) void gemm_f32_wmma(
    const float* __restrict__ A, const float* __restrict__ B,
    const float* __restrict__ bias, float* __restrict__ C,
    int M, int Ncols, int K, int act) {
  __shared__ float As[16][KC + 4];     // pad -> conflict-free column reads
  __shared__ float Bs[KC][128];

  const int row0    = blockIdx.x * 16;
  const int colBase = blockIdx.y * 128;
  const int wave    = threadIdx.x >> 5;
  const int lane    = threadIdx.x & 31;
  const int wcol    = colBase + wave * 16;   // this wave's first output col
  const int half    = lane >> 4;             // 0: lanes 0-15, 1: lanes 16-31
  const int l16     = lane & 15;

  v8f c = {};

  for (int k0 = 0; k0 < K; k0 += KC) {
    // Stage A tile 16 x KC (coalesced: 4 elems/thread)
    for (int t = threadIdx.x; t < 16 * KC; t += 256) {
      int r = t / KC, kk = t % KC;
      int row = row0 + r;
      As[r][kk] = (row < M && (k0 + kk) < K) ? A[(size_t)row * K + k0 + kk] : 0.0f;
    }
    // Stage B tile KC x 128 (coalesced: 32 elems/thread)
    for (int t = threadIdx.x; t < KC * 128; t += 256) {
      int kk = t >> 7, j = t & 127;
      int col = colBase + j;
      Bs[kk][j] = (col < Ncols && (k0 + kk) < K) ? B[(size_t)(k0 + kk) * Ncols + col] : 0.0f;
    }
    __syncthreads();

    if (wcol < Ncols) {
      // 16 x 16x16x4 fp32 WMMA steps per KC chunk.
      for (int kk = 0; kk < KC; kk += 4) {
        // A frag (16x4): VGPR0 = {K=kk | K=kk+2}, VGPR1 = {K=kk+1 | K=kk+3}
        v2f a, b;
        a.x = As[l16][kk + 2 * half];
        a.y = As[l16][kk + 2 * half + 1];
        // B frag (4x16): row striped across lanes, mirrored halves
        b.x = Bs[kk + 2 * half][wave * 16 + l16];
        b.y = Bs[kk + 2 * half + 1][wave * 16 + l16];
        c = __builtin_amdgcn_wmma_f32_16x16x4_f32(
            /*neg_a=*/false, a, /*neg_b=*/false, b,
            /*c_mod=*/(short)0, c, /*reuse_a=*/false, /*reuse_b=*/false);
      }
    }
    __syncthreads();
  }

  // C/D layout: VGPR r -> row r (lanes 0-15) / row r+8 (lanes 16-31)
  if (wcol < Ncols) {
    const int col = wcol + l16;
    for (int r = 0; r < 8; ++r) {
      int row = row0 + r + 8 * half;
      if (row < M) {
        float v = c[r];
        if (bias) v += bias[col];
        if (act)  v = fmaxf(v, 0.0f);
        C[(size_t)row * Ncols + col] = v;
      }
    }
  }
}

// ---------------------------------------------------------------------------
// Small helpers
// ---------------------------------------------------------------------------
__global__ void fill_kernel(float* __restrict__ p, float v, long n) {
  long i = (long)blockIdx.x * blockDim.x + threadIdx.x;
  if (i < n) p[i] = v;
}

__global__ void degree_kernel(const int* __restrict__ dst, float* __restrict__ deg, int E) {
  int e = blockIdx.x * blockDim.x + threadIdx.x;
  if (e < E) atomicAdd(&deg[dst[e]], 1.0f);
}

__global__ void cnt_kernel(const int* __restrict__ batch, float* __restrict__ cnt, int n) {
  int i = blockIdx.x * blockDim.x + threadIdx.x;
  if (i < n) atomicAdd(&cnt[batch[i]], 1.0f);
}

__global__ void rsqrt_kernel(float* __restrict__ deg, int n) {
  int i = blockIdx.x * blockDim.x + threadIdx.x;
  if (i < n) deg[i] = rsqrtf(deg[i]);     // deg >= 1 always (self loop)
}

// acc[i,:] = h[i,:] * dis[i]^2  (self-loop contribution, also inits acc)
__global__ void selfloop_kernel(const float* __restrict__ h, const float* __restrict__ dis,
                                float* __restrict__ acc, int n) {
  long gid = (long)blockIdx.x * blockDim.x + threadIdx.x;
  if (gid >= (long)n * HDIM) return;
  int i = (int)(gid >> 7);
  float di = dis[i];
  acc[gid] = h[gid] * di * di;
}

// One edge per wave32: lane handles a float4 chunk of the 128-wide feature.
__global__ void edge_scatter_kernel(const float* __restrict__ h, const float* __restrict__ dis,
                                    const int* __restrict__ src, const int* __restrict__ dst,
                                    float* __restrict__ acc, int E) {
  long gid = (long)blockIdx.x * blockDim.x + threadIdx.x;
  int e    = (int)(gid >> 5);
  int lane = (int)(gid & 31);
  if (e >= E) return;
  int s = src[e];
  int d = dst[e];
  float w = dis[s] * dis[d];
  const float4 m = ((const float4*)(h + (size_t)s * HDIM))[lane];
  float* p = acc + (size_t)d * HDIM + lane * 4;
  atomicAdd(p + 0, m.x * w);
  atomicAdd(p + 1, m.y * w);
  atomicAdd(p + 2, m.z * w);
  atomicAdd(p + 3, m.w * w);
}

// v = relu(acc + bias); optionally store; pool into g[b, colOff + j] (sum) and
// g[b, colOff + 128 + j] (max via uint bits -- valid since v >= 0, init 0).
__global__ void finish_pool_kernel(const float* __restrict__ acc, const float* __restrict__ bias,
                                   const int* __restrict__ batch, float* __restrict__ x_store,
                                   float* __restrict__ g, int colOff, int n) {
  long gid = (long)blockIdx.x * blockDim.x + threadIdx.x;
  if (gid >= (long)n * HDIM) return;
  int i = (int)(gid >> 7), j = (int)(gid & 127);
  float v = fmaxf(acc[gid] + bias[j], 0.0f);
  if (x_store) x_store[gid] = v;
  int b = batch[i];
  float* gb = g + (size_t)b * (4 * HDIM) + colOff;
  atomicAdd(gb + j, v);
  atomicMax((unsigned int*)(gb + HDIM + j), __float_as_uint(v));
}

// mean = sum / max(cnt,1) for both pooled mean blocks (cols [0,128) & [256,384))
__global__ void mean_div_kernel(float* __restrict__ g, const float* __restrict__ cnt, int G) {
  int gid = blockIdx.x * blockDim.x + threadIdx.x;
  if (gid >= G * HDIM) return;
  int b = gid >> 7, j = gid & 127;
  float inv = 1.0f / fmaxf(cnt[b], 1.0f);
  g[(size_t)b * 512 + j]       *= inv;
  g[(size_t)b * 512 + 256 + j] *= inv;
}

// ---------------------------------------------------------------------------
extern "C" void kernel_launch(void* const* d_in, const int* in_sizes, int n_in,
                              void* d_out, int out_size, void* d_ws, size_t ws_size,
                              hipStream_t stream) {
  const float* x     = (const float*)d_in[0];
  const int*   eidx  = (const int*)d_in[1];      // [2, E]
  const int*   batch = (const int*)d_in[2];
  const float* W1    = (const float*)d_in[3];
  const float* b1    = (const float*)d_in[4];
  const float* W2    = (const float*)d_in[5];
  const float* b2    = (const float*)d_in[6];
  const float* lW1   = (const float*)d_in[7];    // [512, 256]
  const float* lb1   = (const float*)d_in[8];
  const float* lW2   = (const float*)d_in[9];    // [256, 16]
  const float* lb2   = (const float*)d_in[10];

  const int n  = in_sizes[0] / HDIM;             // 50000
  const int E  = in_sizes[1] / 2;                // 800000
  const int C  = in_sizes[10];                   // 16
  const int G  = out_size / C;                   // 512
  const int H4 = 4 * HDIM;                       // 512
  const int H2 = 2 * HDIM;                       // 256
  const int* src = eidx;
  const int* dst = eidx + E;

  // Workspace layout (floats)
  float* ws     = (float*)d_ws;
  float* dis    = ws;                            // n       (degree -> rsqrt)
  float* cnt    = dis + n;                       // G
  float* bufA   = cnt + G;                       // n*128   (h / x0 / acc2)
  float* bufB   = bufA + (size_t)n * HDIM;       // n*128   (acc1 / h2)
  float* gfeat  = bufB + (size_t)n * HDIM;       // G*512
  float* hidden = gfeat + (size_t)G * H4;        // G*256

  const int T = 256;
  const long nH = (long)n * HDIM;
  dim3 blk(T);

  // --- normalization: deg = 1 (self loop) + scatter ones; dis = rsqrt(deg) ---
  fill_kernel<<<(n + T - 1) / T, blk, 0, stream>>>(dis, 1.0f, n);
  fill_kernel<<<(G + T - 1) / T, blk, 0, stream>>>(cnt, 0.0f, G);
  fill_kernel<<<((long)G * H4 + T - 1) / T, blk, 0, stream>>>(gfeat, 0.0f, (long)G * H4);
  degree_kernel<<<(E + T - 1) / T, blk, 0, stream>>>(dst, dis, E);
  cnt_kernel<<<(n + T - 1) / T, blk, 0, stream>>>(batch, cnt, n);
  rsqrt_kernel<<<(n + T - 1) / T, blk, 0, stream>>>(dis, n);

  // --- conv1: h = x @ W1 (WMMA) ; acc = selfloop + edge scatter ; relu+pool ---
  gemm_f32_wmma<<<dim3((n + 15) / 16, 1), blk, 0, stream>>>(
      x, W1, nullptr, bufA, n, HDIM, HDIM, 0);
  selfloop_kernel<<<(nH + T - 1) / T, blk, 0, stream>>>(bufA, dis, bufB, n);
  edge_scatter_kernel<<<((long)E * 32 + T - 1) / T, blk, 0, stream>>>(
      bufA, dis, src, dst, bufB, E);
  finish_pool_kernel<<<(nH + T - 1) / T, blk, 0, stream>>>(
      bufB, b1, batch, bufA /*store x0*/, gfeat, 0, n);

  // --- conv2: h2 = x0 @ W2 ; acc2 = selfloop + scatter ; relu+pool (no store) ---
  gemm_f32_wmma<<<dim3((n + 15) / 16, 1), blk, 0, stream>>>(
      bufA, W2, nullptr, bufB, n, HDIM, HDIM, 0);
  selfloop_kernel<<<(nH + T - 1) / T, blk, 0, stream>>>(bufB, dis, bufA, n);
  edge_scatter_kernel<<<((long)E * 32 + T - 1) / T, blk, 0, stream>>>(
      bufB, dis, src, dst, bufA, E);
  finish_pool_kernel<<<(nH + T - 1) / T, blk, 0, stream>>>(
      bufA, b2, batch, nullptr, gfeat, 2 * HDIM, n);

  // --- pooled means ---
  mean_div_kernel<<<(G * HDIM + T - 1) / T, blk, 0, stream>>>(gfeat, cnt, G);

  // --- MLP head (WMMA, fused bias+relu) ---
  gemm_f32_wmma<<<dim3(G / 16, (H2 + 127) / 128), blk, 0, stream>>>(
      gfeat, lW1, lb1, hidden, G, H2, H4, 1);
  gemm_f32_wmma<<<dim3(G / 16, 1), blk, 0, stream>>>(
      hidden, lW2, lb2, (float*)d_out, G, C, H2, 0);
}